// PointGroup_80427557585281
// MI455X (gfx1250) — compile-verified
//
#include <hip/hip_runtime.h>
#include <hip/hip_bf16.h>
#include <float.h>

// ---------------------------------------------------------------------------
// PointGroup cluster-normalization, MI455X (gfx1250, wave32).
//
// Phase 1: per-cluster sum/min/max of gathered coords -> mean, cscale, off
//          (8 floats per cluster in d_ws).
// Phase 2: per-point gather of feats row (one wave32 = one 32-float row,
//          single coalesced 128B request) + affine coord transform,
//          writing 35-float output rows. Uses global_prefetch_b8 to cover
//          random-gather latency on the next feature row.
// ---------------------------------------------------------------------------

#define STATS_BLOCK 256
#define PW 4   // points per wave in phase 2

__global__ void pg_cluster_stats(const int* __restrict__ cidx,
                                 const int* __restrict__ coff,
                                 const float* __restrict__ coords,
                                 const float* __restrict__ jitter,
                                 const int* __restrict__ fullscale_p,
                                 const int* __restrict__ scale_p,
                                 float* __restrict__ params)  // 8 floats / cluster
{
    const int c   = blockIdx.x;
    const int tid = threadIdx.x;
    const int beg = coff[c];
    const int end = coff[c + 1];

    float sum[3] = {0.f, 0.f, 0.f};
    float mn[3]  = { FLT_MAX,  FLT_MAX,  FLT_MAX};
    float mx[3]  = {-FLT_MAX, -FLT_MAX, -FLT_MAX};

    for (int j = beg + tid; j < end; j += STATS_BLOCK) {
        const int pid = cidx[2 * j + 1];
        const size_t cb = (size_t)pid * 3;
#pragma unroll
        for (int d = 0; d < 3; ++d) {
            const float v = coords[cb + d];
            sum[d] += v;
            mn[d] = fminf(mn[d], v);
            mx[d] = fmaxf(mx[d], v);
        }
    }

    // wave32 reduction
#pragma unroll
    for (int o = 16; o > 0; o >>= 1) {
#pragma unroll
        for (int d = 0; d < 3; ++d) {
            sum[d] += __shfl_down(sum[d], o, 32);
            mn[d] = fminf(mn[d], __shfl_down(mn[d], o, 32));
            mx[d] = fmaxf(mx[d], __shfl_down(mx[d], o, 32));
        }
    }

    __shared__ float sS[3][STATS_BLOCK / 32];
    __shared__ float sN[3][STATS_BLOCK / 32];
    __shared__ float sX[3][STATS_BLOCK / 32];
    const int lane = tid & 31;
    const int wid  = tid >> 5;
    if (lane == 0) {
#pragma unroll
        for (int d = 0; d < 3; ++d) { sS[d][wid] = sum[d]; sN[d][wid] = mn[d]; sX[d][wid] = mx[d]; }
    }
    __syncthreads();

    if (tid == 0) {
#pragma unroll
        for (int d = 0; d < 3; ++d) { sum[d] = sS[d][0]; mn[d] = sN[d][0]; mx[d] = sX[d][0]; }
        for (int w = 1; w < STATS_BLOCK / 32; ++w) {
#pragma unroll
            for (int d = 0; d < 3; ++d) {
                sum[d] += sS[d][w];
                mn[d] = fminf(mn[d], sN[d][w]);
                mx[d] = fmaxf(mx[d], sX[d][w]);
            }
        }

        const float fs     = (float)(*fullscale_p);
        const float scalef = (float)(*scale_p);
        const float cntf   = fmaxf((float)(end - beg), 1.0f);

        float mean[3], cmin[3], cmax[3];
        float t = 0.0f;
#pragma unroll
        for (int d = 0; d < 3; ++d) {
            mean[d] = sum[d] / cntf;
            // min(x - m) == min(x) - m (fp subtraction is monotone): exact.
            cmin[d] = mn[d] - mean[d];
            cmax[d] = mx[d] - mean[d];
            t = fmaxf(t, (cmax[d] - cmin[d]) / fs);
        }
        float cscale = 1.0f / t - 0.01f;
        cscale = fminf(cscale, scalef);

        float* P = params + (size_t)c * 8;
#pragma unroll
        for (int d = 0; d < 3; ++d) {
            const float min_xyz = cmin[d] * cscale;
            const float max_xyz = cmax[d] * cscale;
            const float rng     = max_xyz - min_xyz;
            const float off = -min_xyz
                + fmaxf(fs - rng - 0.001f, 0.0f) * jitter[d]
                + fminf(fs - rng + 0.001f, 0.0f) * jitter[3 + d];
            P[d]     = mean[d];
            P[4 + d] = off;
        }
        P[3] = cscale;
        P[7] = 0.0f;
    }
}

__global__ void pg_gather_out(const int* __restrict__ cidx,
                              const float* __restrict__ feats,
                              const float* __restrict__ coords,
                              const float* __restrict__ params,
                              float* __restrict__ out,
                              int M, int C)
{
    const int wave = (int)((blockIdx.x * blockDim.x + threadIdx.x) >> 5);
    const int lane = threadIdx.x & 31;
    const int i0   = wave * PW;
    const int Cout = C + 3;

#pragma unroll
    for (int p = 0; p < PW; ++p) {
        const int i = i0 + p;
        if (i >= M) return;                    // wave-uniform exit
        const int seg = cidx[2 * i];
        const int pid = cidx[2 * i + 1];

        // Prefetch next point's feature row into cache (global_prefetch_b8).
        if (p + 1 < PW && i + 1 < M) {
            const int pidn = cidx[2 * i + 3];
            __builtin_prefetch(&feats[(size_t)pidn * C + lane], 0, 1);
        }

        const size_t ob = (size_t)i * Cout;
        // One wave32 covers a 32-float feature row: one 128B coalesced load,
        // 128 contiguous bytes of the output row.
        for (int j = lane; j < C; j += 32) {
            out[ob + j] = feats[(size_t)pid * C + j];
        }

        if (lane < 3) {
            const float x    = coords[(size_t)pid * 3 + lane];
            const float mean = params[(size_t)seg * 8 + lane];
            const float s    = params[(size_t)seg * 8 + 3];
            const float off  = params[(size_t)seg * 8 + 4 + lane];
            // Match reference: separate mul then add (no fma contraction).
            const float cc = __fadd_rn(__fmul_rn(__fsub_rn(x, mean), s), off);
            out[ob + C + lane] = cc;
        }
    }
}

extern "C" void kernel_launch(void* const* d_in, const int* in_sizes, int n_in,
                              void* d_out, int out_size, void* d_ws, size_t ws_size,
                              hipStream_t stream)
{
    const int*   cidx      = (const int*)d_in[0];    // (M, 2) int32
    const int*   coff      = (const int*)d_in[1];    // (nC+1,) int32
    const float* feats     = (const float*)d_in[2];  // (N, C) f32
    const float* coords    = (const float*)d_in[3];  // (N, 3) f32
    const float* jitter    = (const float*)d_in[4];  // (2, 3) f32
    const int*   fullscale = (const int*)d_in[5];    // scalar
    const int*   scale     = (const int*)d_in[6];    // scalar
    float*       out       = (float*)d_out;

    const int M  = in_sizes[0] / 2;
    const int nC = in_sizes[1] - 1;
    const int N  = in_sizes[3] / 3;
    const int C  = (N > 0) ? (in_sizes[2] / N) : 32;

    float* params = (float*)d_ws;  // 8 floats per cluster

    if (nC > 0) {
        pg_cluster_stats<<<nC, STATS_BLOCK, 0, stream>>>(
            cidx, coff, coords, jitter, fullscale, scale, params);
    }

    const int waves  = (M + PW - 1) / PW;
    const int blocks = (waves + 7) / 8;   // 8 waves (256 threads) per block
    if (blocks > 0) {
        pg_gather_out<<<blocks, 256, 0, stream>>>(
            cidx, feats, coords, params, out, M, C);
    }
}